// PLRNN_45303315038941
// MI455X (gfx1250) — compile-verified
//
#include <hip/hip_runtime.h>
#include <hip/hip_bf16.h>

#define DZ    128
#define DB    32
#define DSIN  32
#define DX    64
#define BATCH 64
#define TSTEPS 4096

#define BREAL 8            // real batches per workgroup
#define NWG   (BATCH/BREAL) // 8 workgroups
#define NTHREADS 256

#define STH 33             // padded theta row stride (words)  -> bank-conflict free
#define SZR 132            // padded z/be row stride (words)   -> bank-conflict free A-frag gathers

typedef __attribute__((ext_vector_type(2))) float v2f;
typedef __attribute__((ext_vector_type(8))) float v8f;

__launch_bounds__(NTHREADS, 1)
__global__ void plrnn_fused_kernel(const float* __restrict__ z0,
                                   const float* __restrict__ inputs,
                                   const float* __restrict__ AW,
                                   const float* __restrict__ h,
                                   const float* __restrict__ thetas,
                                   const float* __restrict__ alphas,
                                   const float* __restrict__ clipping,
                                   const float* __restrict__ C,
                                   const float* __restrict__ Bobs,
                                   float* __restrict__ out)
{
    __shared__ float th_sh [DZ * STH];     // thetas           (padded)
    __shared__ float th2_sh[DZ * STH];     // thetas - clip    (padded)
    __shared__ float z_sh  [16 * SZR];     // state tile, rows 8..15 are zero padding
    __shared__ float be_sh [16 * SZR];     // basis tile, rows 8..15 stay zero
    __shared__ float al_sh [DB];
    __shared__ float ad_sh [DZ];           // diag(AW)
    __shared__ float h_sh  [DZ];
    __shared__ float hc_sh [DZ];           // h + C @ s_t
    __shared__ float s_sh  [DSIN];

    const int tid   = threadIdx.x;
    const int lane  = tid & 31;
    const int wave  = tid >> 5;
    const int m16   = lane & 15;
    const int khalf = lane >> 4;
    const int b0    = blockIdx.x * BREAL;

    // ---------------- one-time on-chip staging ----------------
    for (int idx = tid; idx < DZ * DB; idx += NTHREADS) {
        int i = idx >> 5, j = idx & 31;
        float t = thetas[idx];
        th_sh [i * STH + j] = t;
        th2_sh[i * STH + j] = t - clipping[idx];
    }
    for (int idx = tid; idx < BREAL * DZ; idx += NTHREADS) {
        int m = idx >> 7, k = idx & 127;
        z_sh[m * SZR + k] = z0[(size_t)(b0 + m) * DZ + k];   // real rows
    }
    for (int idx = tid; idx < 8 * DZ; idx += NTHREADS) {
        int m = (idx >> 7) + 8, k = idx & 127;
        z_sh [m * SZR + k] = 0.0f;                           // padded rows
    }
    for (int idx = tid; idx < 16 * DZ; idx += NTHREADS) {
        int m = idx >> 7, k = idx & 127;
        be_sh[m * SZR + k] = 0.0f;                           // rows 8..15 stay 0 forever
    }
    if (tid < DB) al_sh[tid] = alphas[tid];
    if (tid < DZ) { ad_sh[tid] = AW[tid * DZ + tid]; h_sh[tid] = h[tid]; }

    // register-cached C row for the h + C@s vector (threads 0..127)
    float Crow[DSIN];
    if (tid < DZ) {
        #pragma unroll
        for (int j = 0; j < DSIN; ++j) Crow[j] = C[tid * DSIN + j];
    }

    // register-cached B fragments of W^T (zero diagonal) — wave w owns columns [w*16, w*16+16)
    const int ncol = wave * 16 + m16;
    v2f wtB[DZ / 4];
    #pragma unroll
    for (int k = 0; k < DZ / 4; ++k) {
        int kk = 4 * k + 2 * khalf;
        float bx = (ncol == kk    ) ? 0.0f : AW[(size_t)ncol * DZ + kk];
        float by = (ncol == kk + 1) ? 0.0f : AW[(size_t)ncol * DZ + kk + 1];
        v2f b; b.x = bx; b.y = by;
        wtB[k] = b;
    }

    // register-cached B fragments of B_obs — output tile (wave & 3)
    const int nout = (wave & 3) * 16 + m16;
    v2f boB[DZ / 4];
    #pragma unroll
    for (int k = 0; k < DZ / 4; ++k) {
        int kk = 4 * k + 2 * khalf;
        v2f b;
        b.x = Bobs[(size_t)kk * DX + nout];
        b.y = Bobs[(size_t)(kk + 1) * DX + nout];
        boB[k] = b;
    }
    __syncthreads();

    // ---------------- sequential rollout ----------------
    for (int t = 0; t < TSTEPS; ++t) {
        if (tid < DSIN) s_sh[tid] = inputs[(size_t)t * DSIN + tid];
        __syncthreads();

        // ---- basis expansion: be[b][i] = sum_j a_j*(leaky(z-th) - leaky(z-th2)) ----
        {
            const int i  = tid & 127;
            const int bb = (tid >> 7) * 4;          // 4 batches per thread (2 groups)
            float zr[4], acc[4];
            #pragma unroll
            for (int b = 0; b < 4; ++b) { zr[b] = z_sh[(bb + b) * SZR + i]; acc[b] = 0.0f; }
            #pragma unroll
            for (int j = 0; j < DB; ++j) {
                float th  = th_sh [i * STH + j];
                float th2 = th2_sh[i * STH + j];
                float a   = al_sh[j];
                #pragma unroll
                for (int b = 0; b < 4; ++b) {
                    float d1 = zr[b] - th;
                    float d2 = zr[b] - th2;
                    float l1 = fmaxf(d1, 0.18f * d1);   // LeakyReLU(0.18)
                    float l2 = fmaxf(d2, 0.18f * d2);
                    acc[b] = fmaf(a, l1 - l2, acc[b]);
                }
            }
            #pragma unroll
            for (int b = 0; b < 4; ++b) be_sh[(bb + b) * SZR + i] = acc[b];
        }
        // ---- hc = h + C @ s_t  (fully unrolled: constant Crow indices, no v_movrels) ----
        if (tid < DZ) {
            float a = h_sh[tid];
            #pragma unroll
            for (int j = 0; j < DSIN; ++j) a = fmaf(Crow[j], s_sh[j], a);
            hc_sh[tid] = a;
        }
        __syncthreads();

        // ---- recurrent GEMM: D = be @ W^T  (fp32 WMMA, K=128 in 32 chunks of 4) ----
        {
            v8f d = {0.f,0.f,0.f,0.f,0.f,0.f,0.f,0.f};
            #pragma unroll
            for (int k = 0; k < DZ / 4; ++k) {
                int kk = 4 * k + 2 * khalf;
                v2f a;
                a.x = be_sh[m16 * SZR + kk];
                a.y = be_sh[m16 * SZR + kk + 1];
                d = __builtin_amdgcn_wmma_f32_16x16x4_f32(
                        false, a, false, wtB[k], (short)0, d, false, false);
            }
            // z_new = z * a_diag + D + (h + C s)   on this wave's 16 columns
            const float ad  = ad_sh[ncol];
            const float hcv = hc_sh[ncol];
            #pragma unroll
            for (int v = 0; v < 8; ++v) {
                int m = v + 8 * khalf;             // C/D row layout
                int idx = m * SZR + ncol;
                z_sh[idx] = fmaf(z_sh[idx], ad, d[v] + hcv);
            }
        }
        __syncthreads();

        // ---- fused observation: out rows = z_new @ B_obs (waves 0..3, 4 N-tiles) ----
        if (wave < 4) {
            v8f o = {0.f,0.f,0.f,0.f,0.f,0.f,0.f,0.f};
            #pragma unroll
            for (int k = 0; k < DZ / 4; ++k) {
                int kk = 4 * k + 2 * khalf;
                v2f a;
                a.x = z_sh[m16 * SZR + kk];
                a.y = z_sh[m16 * SZR + kk + 1];
                o = __builtin_amdgcn_wmma_f32_16x16x4_f32(
                        false, a, false, boB[k], (short)0, o, false, false);
            }
            if (khalf == 0) {                      // rows 0..7 are the real batches
                const size_t rowbase = (size_t)t * BATCH + b0;
                #pragma unroll
                for (int v = 0; v < 8; ++v)
                    out[(rowbase + v) * DX + nout] = o[v];
            }
        }
        // next-iteration writes to s_sh/be_sh/z_sh are fenced by the barriers above
    }
}

extern "C" void kernel_launch(void* const* d_in, const int* in_sizes, int n_in,
                              void* d_out, int out_size, void* d_ws, size_t ws_size,
                              hipStream_t stream) {
    const float* z0   = (const float*)d_in[0];
    const float* inp  = (const float*)d_in[1];
    const float* AW   = (const float*)d_in[2];
    const float* hb   = (const float*)d_in[3];
    const float* th   = (const float*)d_in[4];
    const float* al   = (const float*)d_in[5];
    const float* cl   = (const float*)d_in[6];
    const float* C    = (const float*)d_in[7];
    const float* Bo   = (const float*)d_in[8];
    float* out = (float*)d_out;

    plrnn_fused_kernel<<<dim3(NWG), dim3(NTHREADS), 0, stream>>>(
        z0, inp, AW, hb, th, al, cl, C, Bo, out);

    (void)in_sizes; (void)n_in; (void)out_size; (void)d_ws; (void)ws_size;
}